// GATNetwork_24000277250674
// MI455X (gfx1250) — compile-verified
//
#include <hip/hip_runtime.h>
#include <hip/hip_bf16.h>

// ---------------------------------------------------------------------------
// GATv2Conv (H=4 heads, D=32) + global max pool, for MI455X (gfx1250).
//   x:[N,128]  edge_index:[2,E]  batch:[N] (sorted)  W_l/W_r:[128,128]
//   b_l/b_r/bias:[128]  att:[4,32] -> out pooled:[B,128]
// ---------------------------------------------------------------------------

#define Fdim 128      // input features == H*D
#define Hh   4        // heads
#define Dd   32       // per-head dim
#define NEG_SLOPE 0.2f

typedef __attribute__((ext_vector_type(2))) float v2f;
typedef __attribute__((ext_vector_type(8))) float v8f;

// float atomic max via monotone int-pattern trick:
// positive floats order like ints, negative floats order reversed as uints.
// Lowers to single global_atomic_max_i32 / global_atomic_min_u32.
__device__ __forceinline__ void atomicMaxFloat(float* addr, float val) {
    if (val >= 0.0f) {
        atomicMax((int*)addr, __float_as_int(val));
    } else {
        atomicMin((unsigned int*)addr, __float_as_uint(val));
    }
}

// ---------------------------------------------------------------------------
// Kernel 0: initialize workspace + output
// ---------------------------------------------------------------------------
__global__ void k_init(float* __restrict__ agg, float* __restrict__ emax,
                       float* __restrict__ den, float* __restrict__ out,
                       int N, int out_elems) {
    int i = blockIdx.x * blockDim.x + threadIdx.x;
    if (i < N * Fdim) agg[i] = 0.0f;
    if (i < N * Hh) {
        emax[i] = __int_as_float(0xff800000);  // -inf
        den[i]  = 0.0f;
    }
    if (i < out_elems) out[i] = 0.0f;
}

// ---------------------------------------------------------------------------
// Kernel 1: XL = x@W_l + b_l, XR = x@W_r + b_r using V_WMMA_F32_16X16X4_F32.
// One block (8 waves) per 16-row tile; wave w computes column tile w for
// BOTH weight matrices (A fragment shared).
// A 16x4 f32 frag: lane L (m=L&15): a[v] = X[m][k + 2*(L>>4) + v]
// B 4x16  f32 frag: lane L (n=L&15): b[v] = W[k + 2*(L>>4) + v][n]
// C/D 16x16 f32:   VGPR r: row = r + 8*(L>>4), col = L&15
// ---------------------------------------------------------------------------
__global__ void __launch_bounds__(256)
k_gemm_xlxr(const float* __restrict__ x,
            const float* __restrict__ Wl, const float* __restrict__ bl,
            const float* __restrict__ Wr, const float* __restrict__ br,
            float* __restrict__ xl, float* __restrict__ xr, int N) {
    const int lane = threadIdx.x & 31;
    const int nt   = threadIdx.x >> 5;          // column tile 0..7
    const int row0 = blockIdx.x * 16;

    int mrow = row0 + (lane & 15);
    if (mrow >= N) mrow = N - 1;                // clamp: keep EXEC all-ones for WMMA
    const int col   = nt * 16 + (lane & 15);    // 0..127
    const int khalf = (lane >> 4) * 2;          // 0 or 2

    v8f accL = {0.f,0.f,0.f,0.f,0.f,0.f,0.f,0.f};
    v8f accR = {0.f,0.f,0.f,0.f,0.f,0.f,0.f,0.f};

    const float* __restrict__ xrow = x + (size_t)mrow * Fdim;

#pragma unroll 4
    for (int k = 0; k < Fdim; k += 4) {
        const int kk = k + khalf;
        v2f a, bLf, bRf;
        a.x   = xrow[kk];
        a.y   = xrow[kk + 1];
        bLf.x = Wl[(size_t)kk * Fdim + col];
        bLf.y = Wl[(size_t)(kk + 1) * Fdim + col];
        bRf.x = Wr[(size_t)kk * Fdim + col];
        bRf.y = Wr[(size_t)(kk + 1) * Fdim + col];
        accL = __builtin_amdgcn_wmma_f32_16x16x4_f32(
            false, a, false, bLf, (short)0, accL, false, false);
        accR = __builtin_amdgcn_wmma_f32_16x16x4_f32(
            false, a, false, bRf, (short)0, accR, false, false);
    }

    const float blv = bl[col];
    const float brv = br[col];
#pragma unroll
    for (int r = 0; r < 8; ++r) {
        const int orow = row0 + r + (lane >> 4) * 8;
        if (orow < N) {
            xl[(size_t)orow * Fdim + col] = accL[r] + blv;
            xr[(size_t)orow * Fdim + col] = accR[r] + brv;
        }
    }
}

// ---------------------------------------------------------------------------
// Kernel 2: per-edge attention logits e[t,h] = sum_d LeakyReLU(xl_j+xr_i)*att
// One wave per edge. Lane l handles features 4l..4l+3 (head h = l>>3).
// 8-lane shuffle reduction -> per-head logit; atomic max into emax[dst,h].
// Edge t<E: (src,dst)=(ei[t],ei[E+t]); t>=E: self loop (t-E, t-E).
// ---------------------------------------------------------------------------
__global__ void __launch_bounds__(256)
k_edge_logits(const float* __restrict__ xl, const float* __restrict__ xr,
              const int* __restrict__ ei, const float* __restrict__ att,
              float* __restrict__ e_ws, float* __restrict__ emax,
              int E, int N) {
    const int lane   = threadIdx.x & 31;
    const int wid    = (blockIdx.x * blockDim.x + threadIdx.x) >> 5;
    const int nwaves = (gridDim.x * blockDim.x) >> 5;
    const int EN     = E + N;

    const float4 av = *(const float4*)(att + lane * 4);  // loop-invariant

    for (int t = wid; t < EN; t += nwaves) {
        int s, d;
        if (t < E) { s = ei[t]; d = ei[E + t]; }
        else       { s = t - E; d = s; }

        const float4 xlv = *(const float4*)(xl + (size_t)s * Fdim + lane * 4);
        const float4 xrv = *(const float4*)(xr + (size_t)d * Fdim + lane * 4);

        float4 m;
        m.x = xlv.x + xrv.x; m.y = xlv.y + xrv.y;
        m.z = xlv.z + xrv.z; m.w = xlv.w + xrv.w;
        m.x = (m.x > 0.f) ? m.x : NEG_SLOPE * m.x;
        m.y = (m.y > 0.f) ? m.y : NEG_SLOPE * m.y;
        m.z = (m.z > 0.f) ? m.z : NEG_SLOPE * m.z;
        m.w = (m.w > 0.f) ? m.w : NEG_SLOPE * m.w;
        float p = m.x * av.x + m.y * av.y + m.z * av.z + m.w * av.w;

        // reduce over the 8 lanes of one head
        p += __shfl_xor(p, 1);
        p += __shfl_xor(p, 2);
        p += __shfl_xor(p, 4);

        if ((lane & 7) == 0) {
            const int h = lane >> 3;
            e_ws[(size_t)t * Hh + h] = p;
            atomicMaxFloat(emax + (size_t)d * Hh + h, p);
        }
    }
}

// ---------------------------------------------------------------------------
// Kernel 3: softmax denominators. Thread per (edge, head).
// ---------------------------------------------------------------------------
__global__ void __launch_bounds__(256)
k_edge_den(const int* __restrict__ ei, const float* __restrict__ e_ws,
           const float* __restrict__ emax, float* __restrict__ den,
           int E, int N) {
    const int i = blockIdx.x * blockDim.x + threadIdx.x;
    const int EN = E + N;
    if (i >= EN * Hh) return;
    const int t = i >> 2;
    const int h = i & 3;
    const int d = (t < E) ? ei[E + t] : (t - E);
    const float ex = __expf(e_ws[i] - emax[(size_t)d * Hh + h]);
    atomicAdd(den + (size_t)d * Hh + h, ex);
}

// ---------------------------------------------------------------------------
// Kernel 4: aggregate agg[dst] += alpha * xl[src]. One wave per edge.
// ---------------------------------------------------------------------------
__global__ void __launch_bounds__(256)
k_edge_aggregate(const float* __restrict__ xl, const int* __restrict__ ei,
                 const float* __restrict__ e_ws, const float* __restrict__ emax,
                 const float* __restrict__ den, float* __restrict__ agg,
                 int E, int N) {
    const int lane   = threadIdx.x & 31;
    const int wid    = (blockIdx.x * blockDim.x + threadIdx.x) >> 5;
    const int nwaves = (gridDim.x * blockDim.x) >> 5;
    const int EN     = E + N;
    const int h      = lane >> 3;

    for (int t = wid; t < EN; t += nwaves) {
        int s, d;
        if (t < E) { s = ei[t]; d = ei[E + t]; }
        else       { s = t - E; d = s; }

        const float e  = e_ws[(size_t)t * Hh + h];
        const float mx = emax[(size_t)d * Hh + h];
        const float dn = den[(size_t)d * Hh + h];
        const float alpha = __expf(e - mx) / dn;

        const float4 xlv = *(const float4*)(xl + (size_t)s * Fdim + lane * 4);
        float* base = agg + (size_t)d * Fdim + lane * 4;
        atomicAdd(base + 0, alpha * xlv.x);
        atomicAdd(base + 1, alpha * xlv.y);
        atomicAdd(base + 2, alpha * xlv.z);
        atomicAdd(base + 3, alpha * xlv.w);
    }
}

// ---------------------------------------------------------------------------
// Kernel 5: out = relu(agg + bias); pooled[b] = max over nodes of graph b.
// 128 threads (one per feature) per 64-node chunk. batch[] is sorted, so we
// keep a running per-feature max and flush one atomicMax per batch change.
// d_out pre-zeroed; relu >= 0 makes this exactly the reference semantics
// (empty graphs stay 0, matching the isfinite guard).
// ---------------------------------------------------------------------------
#define POOL_NODES 64
__global__ void __launch_bounds__(128)
k_pool(const float* __restrict__ agg, const float* __restrict__ bias,
       const int* __restrict__ batch, float* __restrict__ out, int N) {
    const int f  = threadIdx.x;
    const int n0 = blockIdx.x * POOL_NODES;
    const int n1 = min(n0 + POOL_NODES, N);
    const float bf = bias[f];

    int cb = -1;
    float rmax = 0.0f;
    for (int n = n0; n < n1; ++n) {
        const int b = batch[n];
        float v = agg[(size_t)n * Fdim + f] + bf;
        v = (v > 0.0f) ? v : 0.0f;
        if (b != cb) {
            if (cb >= 0) atomicMaxFloat(out + (size_t)cb * Fdim + f, rmax);
            cb = b;
            rmax = v;
        } else {
            rmax = fmaxf(rmax, v);
        }
    }
    if (cb >= 0) atomicMaxFloat(out + (size_t)cb * Fdim + f, rmax);
}

// ---------------------------------------------------------------------------
extern "C" void kernel_launch(void* const* d_in, const int* in_sizes, int n_in,
                              void* d_out, int out_size, void* d_ws, size_t ws_size,
                              hipStream_t stream) {
    const float* x     = (const float*)d_in[0];
    const int*   ei    = (const int*)  d_in[1];
    const int*   batch = (const int*)  d_in[2];
    const float* W_l   = (const float*)d_in[3];
    const float* b_l   = (const float*)d_in[4];
    const float* W_r   = (const float*)d_in[5];
    const float* b_r   = (const float*)d_in[6];
    const float* att   = (const float*)d_in[7];
    const float* bias  = (const float*)d_in[8];
    float*       out   = (float*)d_out;

    const int N  = in_sizes[0] / Fdim;
    const int E  = in_sizes[1] / 2;
    const int EN = E + N;

    // workspace layout (floats)
    float* xl   = (float*)d_ws;
    float* xr   = xl + (size_t)N * Fdim;
    float* agg  = xr + (size_t)N * Fdim;
    float* e_ws = agg + (size_t)N * Fdim;
    float* emax = e_ws + (size_t)EN * Hh;
    float* den  = emax + (size_t)N * Hh;

    // 0) init agg / emax / den / out
    {
        const int total = N * Fdim;
        k_init<<<(total + 255) / 256, 256, 0, stream>>>(agg, emax, den, out,
                                                        N, out_size);
    }
    // 1) WMMA GEMM: xl, xr
    {
        const int mtiles = (N + 15) / 16;
        k_gemm_xlxr<<<mtiles, 256, 0, stream>>>(x, W_l, b_l, W_r, b_r, xl, xr, N);
    }
    // 2) per-edge logits + segment max
    {
        const int blocks = (EN + 7) / 8;   // 8 waves per 256-thread block
        k_edge_logits<<<blocks, 256, 0, stream>>>(xl, xr, ei, att, e_ws, emax, E, N);
    }
    // 3) softmax denominators
    {
        const int total = EN * Hh;
        k_edge_den<<<(total + 255) / 256, 256, 0, stream>>>(ei, e_ws, emax, den, E, N);
    }
    // 4) weighted scatter aggregation
    {
        const int blocks = (EN + 7) / 8;
        k_edge_aggregate<<<blocks, 256, 0, stream>>>(xl, ei, e_ws, emax, den, agg, E, N);
    }
    // 5) bias + relu + global max pool
    {
        const int blocks = (N + POOL_NODES - 1) / POOL_NODES;
        k_pool<<<blocks, 128, 0, stream>>>(agg, bias, batch, out, N);
    }
}